// CausalGNNCore_18614388261668
// MI455X (gfx1250) — compile-verified
//
#include <hip/hip_runtime.h>
#include <math.h>

// MI455X / gfx1250, wave32. f32 WMMA 16x16x4 used for all HxH projections.
typedef __attribute__((ext_vector_type(2))) float v2f;
typedef __attribute__((ext_vector_type(8))) float v8f;

#define WMMA_F32(a, b, c) \
  __builtin_amdgcn_wmma_f32_16x16x4_f32(false, (a), false, (b), (short)0, (c), false, false)

#if defined(__gfx1250__) && __has_builtin(__builtin_amdgcn_global_load_async_to_lds_b128) && \
    __has_builtin(__builtin_amdgcn_s_wait_asynccnt)
#define HAVE_ASYNC_LDS 1
#else
#define HAVE_ASYNC_LDS 0
#endif

#if HAVE_ASYNC_LDS
// Match the builtin's parameter types exactly (from the compiler diagnostic):
// non-const pointer to generic (gcc-style) int vector of 16 bytes.
typedef int vi4 __attribute__((vector_size(16)));
typedef __attribute__((address_space(1))) vi4* gas_vi4p;   // global (__device__)
typedef __attribute__((address_space(3))) vi4* las_vi4p;   // LDS (__shared__)
#endif

#define Bsz 128
#define Dn  128
#define Hn  32
#define NROWS (Bsz * Dn)   // 16384

// ---------------------------------------------------------------------------
// colsum[i] = sum_{j != i} W[j, i]
__global__ void k_colsum(const float* __restrict__ W, float* __restrict__ colsum) {
  const int i = threadIdx.x;
  float s = 0.f;
  for (int j = 0; j < Dn; ++j) s += W[j * Dn + i];
  s -= W[i * Dn + i];
  colsum[i] = s;
}

// ---------------------------------------------------------------------------
// Fused: h = tanh(X*enc_w + enc_b) computed on the fly (never stored),
//   hs = h @ m1_w[0:32,:]
//   hd = h @ m1_w[32:64,:] + m1_b
//   ho = h @ o1_w[0:32,:]
// One wave per 16-row tile; K=32 in 8 WMMA steps; N=32 in 2 tiles -> 48 WMMAs.
__global__ __launch_bounds__(256) void k_enc_mm(
    const float* __restrict__ X, const float* __restrict__ enc_w,
    const float* __restrict__ enc_b, const float* __restrict__ m1_w,
    const float* __restrict__ m1_b, const float* __restrict__ o1_w,
    float* __restrict__ hs, float* __restrict__ hd, float* __restrict__ ho) {
  const int tid   = threadIdx.x;
  const int lane  = tid & 31;
  const int wave  = tid >> 5;
  const int rowbase = (blockIdx.x * 8 + wave) * 16;
  const int lm = lane & 15, half = lane >> 4;

  // warm the weight cachelines (global_prefetch_b8)
  __builtin_prefetch(m1_w + lane * 16, 0, 3);   // 64x32 f32 = 8KB / 32 lanes
  __builtin_prefetch(o1_w + lane * 16, 0, 3);

  const float x = X[rowbase + lm];  // A-fragment row for this lane

  v8f hs0 = {}, hs1 = {}, hd0 = {}, hd1 = {}, ho0 = {}, ho1 = {};

#pragma unroll
  for (int k0 = 0; k0 < 32; k0 += 4) {
    const int k = k0 + 2 * half;   // lane's K pair within this 16x4 step
    v2f a, b;
    a.x = tanhf(x * enc_w[k] + enc_b[k]);
    a.y = tanhf(x * enc_w[k + 1] + enc_b[k + 1]);

    // hs: m1_w rows 0..31
    b.x = m1_w[k * Hn + lm];            b.y = m1_w[(k + 1) * Hn + lm];
    hs0 = WMMA_F32(a, b, hs0);
    b.x = m1_w[k * Hn + 16 + lm];       b.y = m1_w[(k + 1) * Hn + 16 + lm];
    hs1 = WMMA_F32(a, b, hs1);
    // hd: m1_w rows 32..63
    b.x = m1_w[(32 + k) * Hn + lm];     b.y = m1_w[(33 + k) * Hn + lm];
    hd0 = WMMA_F32(a, b, hd0);
    b.x = m1_w[(32 + k) * Hn + 16 + lm];b.y = m1_w[(33 + k) * Hn + 16 + lm];
    hd1 = WMMA_F32(a, b, hd1);
    // ho: o1_w rows 0..31
    b.x = o1_w[k * Hn + lm];            b.y = o1_w[(k + 1) * Hn + lm];
    ho0 = WMMA_F32(a, b, ho0);
    b.x = o1_w[k * Hn + 16 + lm];       b.y = o1_w[(k + 1) * Hn + 16 + lm];
    ho1 = WMMA_F32(a, b, ho1);
  }

  const float mb0 = m1_b[lm], mb1 = m1_b[16 + lm];
#pragma unroll
  for (int e = 0; e < 8; ++e) {        // D layout: row = e + 8*half, col = lm (+16)
    const int row = rowbase + e + 8 * half;
    hs[row * Hn + lm]      = hs0[e];
    hs[row * Hn + 16 + lm] = hs1[e];
    hd[row * Hn + lm]      = hd0[e] + mb0;   // fold m1_b into hd
    hd[row * Hn + 16 + lm] = hd1[e] + mb1;
    ho[row * Hn + lm]      = ho0[e];
    ho[row * Hn + 16 + lm] = ho1[e];
  }
}

// ---------------------------------------------------------------------------
// r[b,i,h] = sum_{j != i} W[j,i] * relu(hs[b,j,h] + hd[b,i,h])
// Block = 8 waves, one (b, i) per wave, lane = h. hs[b] (16KB) staged in LDS
// via async global->LDS DMA (ASYNCcnt) when available; each wave stages its
// W column (diagonal zeroed) in LDS for broadcast reads, overlapped with DMA.
__global__ __launch_bounds__(256) void k_agg(
    const float* __restrict__ hs, const float* __restrict__ hd,
    const float* __restrict__ W, float* __restrict__ r) {
  __shared__ float hsld[Dn * Hn];      // 16 KB
  __shared__ float wcol[8][Dn];        // 4 KB

  const int tid  = threadIdx.x;
  const int b    = blockIdx.x >> 4;
  const int igrp = blockIdx.x & 15;

  // stage hs[b]: 4096 floats = 1024 x b128, 256 lanes x 4 ops
#if HAVE_ASYNC_LDS
  {
    const char* gsrc = (const char*)(hs + (size_t)b * Dn * Hn);
#pragma unroll
    for (int t = 0; t < 4; ++t) {
      const size_t off = (size_t)(tid + t * 256) * 16;
      gas_vi4p g = (gas_vi4p)(uintptr_t)(gsrc + off);
      las_vi4p l = (las_vi4p)(uint32_t)(uintptr_t)((char*)hsld + off);
      __builtin_amdgcn_global_load_async_to_lds_b128(g, l, 0, 0);
    }
  }
#else
  {
    const float4* src = (const float4*)(hs + (size_t)b * Dn * Hn);
    float4* dst = (float4*)hsld;
#pragma unroll
    for (int t = 0; t < 4; ++t) dst[tid + t * 256] = src[tid + t * 256];
  }
#endif

  const int wave = tid >> 5, lane = tid & 31;
  const int i = igrp * 8 + wave;
#pragma unroll
  for (int q = 0; q < 4; ++q) {        // W column i, diag zeroed (overlaps DMA)
    const int j = lane + q * 32;
    const float wv = W[j * Dn + i];
    wcol[wave][j] = (j == i) ? 0.f : wv;
  }
  const float hdv = hd[((size_t)b * Dn + i) * Hn + lane];

#if HAVE_ASYNC_LDS
  __builtin_amdgcn_s_wait_asynccnt(0);
#endif
  __syncthreads();

  float acc = 0.f;
#pragma unroll 8
  for (int j = 0; j < Dn; ++j) {
    acc = fmaf(wcol[wave][j], fmaxf(hsld[j * Hn + lane] + hdv, 0.f), acc);
  }
  r[((size_t)b * Dn + i) * Hn + lane] = acc;
}

// ---------------------------------------------------------------------------
// agg = r @ m2_w + colsum[i] * m2_b          (WMMA, D-layout regs)
//   -> LDS transpose to A-layout ->
// z = relu(ho + agg @ o1_w[32:64,:] + o1_b)  (WMMA, C initialized from ho)
// out = z @ o2_w + o2_b                      (shfl_xor 16-lane reduction)
__global__ __launch_bounds__(256) void k_final(
    const float* __restrict__ r, const float* __restrict__ m2_w,
    const float* __restrict__ m2_b, const float* __restrict__ colsum,
    const float* __restrict__ ho, const float* __restrict__ o1_w,
    const float* __restrict__ o1_b, const float* __restrict__ o2_w,
    const float* __restrict__ o2_b, float* __restrict__ out) {
  __shared__ float trans[8][16 * Hn];  // 16 KB, one 16x32 tile per wave

  const int tid  = threadIdx.x;
  const int lane = tid & 31, wave = tid >> 5;
  const int rowbase = (blockIdx.x * 8 + wave) * 16;
  const int lm = lane & 15, half = lane >> 4;

  __builtin_prefetch(m2_w + lane * 32, 0, 3);           // 32x32 f32 = 4KB
  __builtin_prefetch(o1_w + 32 * Hn + lane * 32, 0, 3); // o1_w rows 32..63

  // --- agg = r @ m2_w ---
  v8f a0 = {}, a1 = {};
#pragma unroll
  for (int k0 = 0; k0 < 32; k0 += 4) {
    const int k = k0 + 2 * half;
    v2f a, b;
    a.x = r[(size_t)(rowbase + lm) * Hn + k];
    a.y = r[(size_t)(rowbase + lm) * Hn + k + 1];
    b.x = m2_w[k * Hn + lm];       b.y = m2_w[(k + 1) * Hn + lm];
    a0 = WMMA_F32(a, b, a0);
    b.x = m2_w[k * Hn + 16 + lm];  b.y = m2_w[(k + 1) * Hn + 16 + lm];
    a1 = WMMA_F32(a, b, a1);
  }

  // + colsum[i] * m2_b, then spill tile to LDS for layout change
  const float m2b0 = m2_b[lm], m2b1 = m2_b[16 + lm];
#pragma unroll
  for (int e = 0; e < 8; ++e) {
    const int row = rowbase + e + 8 * half;
    const float cs = colsum[row & (Dn - 1)];  // i = row % 128
    a0[e] += cs * m2b0;
    a1[e] += cs * m2b1;
    trans[wave][(e + 8 * half) * Hn + lm]      = a0[e];
    trans[wave][(e + 8 * half) * Hn + 16 + lm] = a1[e];
  }
  __syncthreads();

  // --- z = ho + agg @ o1_w[32:64,:] ---  (C operand seeded with ho)
  v8f z0, z1;
#pragma unroll
  for (int e = 0; e < 8; ++e) {
    const int row = rowbase + e + 8 * half;
    z0[e] = ho[(size_t)row * Hn + lm];
    z1[e] = ho[(size_t)row * Hn + 16 + lm];
  }
#pragma unroll
  for (int k0 = 0; k0 < 32; k0 += 4) {
    const int k = k0 + 2 * half;
    v2f a, b;
    a.x = trans[wave][lm * Hn + k];
    a.y = trans[wave][lm * Hn + k + 1];
    b.x = o1_w[(32 + k) * Hn + lm];       b.y = o1_w[(33 + k) * Hn + lm];
    z0 = WMMA_F32(a, b, z0);
    b.x = o1_w[(32 + k) * Hn + 16 + lm];  b.y = o1_w[(33 + k) * Hn + 16 + lm];
    z1 = WMMA_F32(a, b, z1);
  }

  // --- out = relu(z + o1_b) @ o2_w + o2_b ---
  const float ob0 = o1_b[lm], ob1 = o1_b[16 + lm];
  const float w0 = o2_w[lm], w1 = o2_w[16 + lm];
  float pv[8];
#pragma unroll
  for (int e = 0; e < 8; ++e) {
    const float za = fmaxf(z0[e] + ob0, 0.f);
    const float zb = fmaxf(z1[e] + ob1, 0.f);
    pv[e] = za * w0 + zb * w1;
  }
#pragma unroll
  for (int m = 1; m < 16; m <<= 1) {   // reduce across the 16 lanes of each half
#pragma unroll
    for (int e = 0; e < 8; ++e) pv[e] += __shfl_xor(pv[e], m, 32);
  }
  if (lm == 0) {
    const float b2 = o2_b[0];
#pragma unroll
    for (int e = 0; e < 8; ++e) out[rowbase + e + 8 * half] = pv[e] + b2;
  }
}

// ---------------------------------------------------------------------------
extern "C" void kernel_launch(void* const* d_in, const int* in_sizes, int n_in,
                              void* d_out, int out_size, void* d_ws, size_t ws_size,
                              hipStream_t stream) {
  const float* X     = (const float*)d_in[0];
  const float* W     = (const float*)d_in[1];
  const float* enc_w = (const float*)d_in[2];
  const float* enc_b = (const float*)d_in[3];
  const float* m1_w  = (const float*)d_in[4];
  const float* m1_b  = (const float*)d_in[5];
  const float* m2_w  = (const float*)d_in[6];
  const float* m2_b  = (const float*)d_in[7];
  const float* o1_w  = (const float*)d_in[8];
  const float* o1_b  = (const float*)d_in[9];
  const float* o2_w  = (const float*)d_in[10];
  const float* o2_b  = (const float*)d_in[11];

  float* ws     = (float*)d_ws;
  float* hs     = ws;                       // 16384*32 f32 = 2 MB
  float* hd     = ws + (size_t)NROWS * Hn;  // 2 MB
  float* ho     = ws + 2 * (size_t)NROWS * Hn;
  float* rbuf   = ws + 3 * (size_t)NROWS * Hn;
  float* colsum = ws + 4 * (size_t)NROWS * Hn;  // 128 f32
  float* out = (float*)d_out;

  k_colsum<<<1, 128, 0, stream>>>(W, colsum);
  k_enc_mm<<<NROWS / (16 * 8), 256, 0, stream>>>(X, enc_w, enc_b, m1_w, m1_b,
                                                 o1_w, hs, hd, ho);
  k_agg<<<Bsz * (Dn / 8), 256, 0, stream>>>(hs, hd, W, rbuf);
  k_final<<<NROWS / (16 * 8), 256, 0, stream>>>(rbuf, m2_w, m2_b, colsum, ho,
                                                o1_w, o1_b, o2_w, o2_b, out);
}